// LinearSelfAttention_69810398429420
// MI455X (gfx1250) — compile-verified
//
#include <hip/hip_runtime.h>

typedef __attribute__((ext_vector_type(2))) float v2f;
typedef __attribute__((ext_vector_type(8))) float v8f;

#define BATCH 4
#define HEADS 16
#define SEQ   4096
#define DIM   64
#define NHEAD (BATCH * HEADS)          // 64 (b,h) pairs
#define KVELEMS (NHEAD * DIM * DIM)    // 262144 floats = 1 MiB state

// ---------------------------------------------------------------------------
// Kernel 0: zero the kv state in workspace (ws is poisoned by the harness).
// ---------------------------------------------------------------------------
__global__ void lsa_zero_kv(float* __restrict__ kvws) {
    int idx = (blockIdx.x * blockDim.x + threadIdx.x) * 4;
    if (idx < KVELEMS) {
        kvws[idx + 0] = 0.0f;
        kvws[idx + 1] = 0.0f;
        kvws[idx + 2] = 0.0f;
        kvws[idx + 3] = 0.0f;
    }
}

// ---------------------------------------------------------------------------
// Kernel 1: kv[h] = K[h]^T @ V[h]   (64x64 per head, accumulated over S)
// Grid: NHEAD * 4 blocks, 256 threads (8 waves).
// Wave w of a block: d-tile = (w&3)*16, s-half = w>>2 (512 rows each).
// Each wave holds 4 f32 WMMA accumulators covering kv[d0:d0+16, 0:64] and
// reduces into workspace with global f32 atomics.
// ---------------------------------------------------------------------------
__global__ void lsa_phase1(const float* __restrict__ K,
                           const float* __restrict__ V,
                           float* __restrict__ kvws) {
    const int SCHUNKS = 4;                     // blocks per head
    int head  = blockIdx.x / SCHUNKS;
    int chunk = blockIdx.x % SCHUNKS;

    int wave = threadIdx.x >> 5;
    int lane = threadIdx.x & 31;

    int d0    = (wave & 3) * 16;               // kv row tile
    int shalf = wave >> 2;                     // 0 or 1
    int s_begin = chunk * (SEQ / SCHUNKS) + shalf * (SEQ / SCHUNKS / 2);
    int s_end   = s_begin + (SEQ / SCHUNKS / 2);   // 512 rows per wave

    const float* __restrict__ Kp = K + (size_t)head * SEQ * DIM;
    const float* __restrict__ Vp = V + (size_t)head * SEQ * DIM;

    int row = lane & 15;                       // M (for A) / N (for B)
    int kb  = (lane >> 4) * 2;                 // K sub-offset per ISA layout

    v8f acc0 = {}; v8f acc1 = {}; v8f acc2 = {}; v8f acc3 = {};

    for (int s = s_begin; s < s_end; s += 4) {
        // A = K^T tile: A[m][k] = K[s+k][d0+m]
        v2f a;
        a.x = Kp[(size_t)(s + kb    ) * DIM + d0 + row];
        a.y = Kp[(size_t)(s + kb + 1) * DIM + d0 + row];

        // B = V tile: B[k][n] = V[s+k][e0+n], four N-tiles e0 = 0,16,32,48
        const float* v0 = Vp + (size_t)(s + kb    ) * DIM + row;
        const float* v1 = Vp + (size_t)(s + kb + 1) * DIM + row;
        v2f b0; b0.x = v0[ 0]; b0.y = v1[ 0];
        v2f b1; b1.x = v0[16]; b1.y = v1[16];
        v2f b2; b2.x = v0[32]; b2.y = v1[32];
        v2f b3; b3.x = v0[48]; b3.y = v1[48];

        acc0 = __builtin_amdgcn_wmma_f32_16x16x4_f32(false, a, false, b0,
                                                     (short)0, acc0, false, false);
        acc1 = __builtin_amdgcn_wmma_f32_16x16x4_f32(false, a, false, b1,
                                                     (short)0, acc1, false, false);
        acc2 = __builtin_amdgcn_wmma_f32_16x16x4_f32(false, a, false, b2,
                                                     (short)0, acc2, false, false);
        acc3 = __builtin_amdgcn_wmma_f32_16x16x4_f32(false, a, false, b3,
                                                     (short)0, acc3, false, false);
    }

    // Reduce partial kv into workspace. C/D layout: VGPR v holds
    // M = v + 8*(lane>=16), N = lane&15.
    float* kvh = kvws + (size_t)head * DIM * DIM;
    int mofs = 8 * (lane >> 4);
#pragma unroll
    for (int v = 0; v < 8; ++v) {
        int d = d0 + v + mofs;
        unsafeAtomicAdd(&kvh[d * DIM + 0 * 16 + row], acc0[v]);
        unsafeAtomicAdd(&kvh[d * DIM + 1 * 16 + row], acc1[v]);
        unsafeAtomicAdd(&kvh[d * DIM + 2 * 16 + row], acc2[v]);
        unsafeAtomicAdd(&kvh[d * DIM + 3 * 16 + row], acc3[v]);
    }
}

// ---------------------------------------------------------------------------
// Kernel 2: out[h] = Q[h] @ kv[h]   (4096x64 · 64x64 per head)
// One wave per 16 Q-rows; 4 accumulators cover N = 0..63; K=64 in 16 steps.
// ---------------------------------------------------------------------------
__global__ void lsa_phase2(const float* __restrict__ Q,
                           const float* __restrict__ kvws,
                           float* __restrict__ out) {
    int wave = threadIdx.x >> 5;
    int lane = threadIdx.x & 31;
    int gwave = blockIdx.x * (blockDim.x >> 5) + wave;

    int head   = gwave >> 8;                 // 256 row-chunks per head
    int mchunk = gwave & 255;
    int m0 = mchunk * 16;

    const float* __restrict__ Qp  = Q + ((size_t)head * SEQ + m0) * DIM;
    const float* __restrict__ kvh = kvws + (size_t)head * DIM * DIM;
    float* __restrict__ op        = out + ((size_t)head * SEQ + m0) * DIM;

    int row = lane & 15;
    int kb  = (lane >> 4) * 2;

    v8f acc0 = {}; v8f acc1 = {}; v8f acc2 = {}; v8f acc3 = {};

#pragma unroll 4
    for (int kk = 0; kk < 16; ++kk) {
        int k0 = kk * 4;
        // A[m][k] = Q[m0+m][k0+k]; per-lane 8B-contiguous pair
        const float* ap = Qp + row * DIM + k0 + kb;
        v2f a; a.x = ap[0]; a.y = ap[1];

        // B[k][n] = kv[k0+k][n]
        const float* b0p = kvh + (k0 + kb    ) * DIM + row;
        const float* b1p = kvh + (k0 + kb + 1) * DIM + row;
        v2f b0; b0.x = b0p[ 0]; b0.y = b1p[ 0];
        v2f b1; b1.x = b0p[16]; b1.y = b1p[16];
        v2f b2; b2.x = b0p[32]; b2.y = b1p[32];
        v2f b3; b3.x = b0p[48]; b3.y = b1p[48];

        acc0 = __builtin_amdgcn_wmma_f32_16x16x4_f32(false, a, false, b0,
                                                     (short)0, acc0, false, false);
        acc1 = __builtin_amdgcn_wmma_f32_16x16x4_f32(false, a, false, b1,
                                                     (short)0, acc1, false, false);
        acc2 = __builtin_amdgcn_wmma_f32_16x16x4_f32(false, a, false, b2,
                                                     (short)0, acc2, false, false);
        acc3 = __builtin_amdgcn_wmma_f32_16x16x4_f32(false, a, false, b3,
                                                     (short)0, acc3, false, false);
    }

    int mofs = 8 * (lane >> 4);
#pragma unroll
    for (int v = 0; v < 8; ++v) {
        float* r = op + (v + mofs) * DIM + row;
        r[ 0] = acc0[v];
        r[16] = acc1[v];
        r[32] = acc2[v];
        r[48] = acc3[v];
    }
}

// ---------------------------------------------------------------------------
extern "C" void kernel_launch(void* const* d_in, const int* in_sizes, int n_in,
                              void* d_out, int out_size, void* d_ws, size_t ws_size,
                              hipStream_t stream) {
    const float* q = (const float*)d_in[0];
    const float* k = (const float*)d_in[1];
    const float* v = (const float*)d_in[2];
    float* out = (float*)d_out;
    float* kvws = (float*)d_ws;   // NHEAD * 64 * 64 floats = 1 MiB

    // 1) zero kv state
    lsa_zero_kv<<<KVELEMS / (256 * 4), 256, 0, stream>>>(kvws);

    // 2) kv = K^T V per head (64 heads x 4 S-chunks)
    lsa_phase1<<<NHEAD * 4, 256, 0, stream>>>(k, v, kvws);

    // 3) out = Q kv   (16384 waves, 8 waves/block)
    lsa_phase2<<<(NHEAD * (SEQ / 16)) / 8, 256, 0, stream>>>(q, kvws, out);
}